// NET_77532749627674
// MI455X (gfx1250) — compile-verified
//
#include <hip/hip_runtime.h>
#include <hip/hip_bf16.h>

#define D 128
#define TILE 32            // rows (edges or nodes) per block
#define A_STRIDE (D + 4)   // LDS row stride in floats (pad to avoid bank conflicts)
#define PACK_FLOATS 16384  // 8 colTiles * 16 kQuads * 32 lanes * 4 floats (= 64KB, per matrix)

typedef __attribute__((ext_vector_type(2))) float v2f;
typedef __attribute__((ext_vector_type(8))) float v8f;

static constexpr float RRELU_SLOPE = (1.0f / 8.0f + 1.0f / 3.0f) * 0.5f;

// ---------------------------------------------------------------------------
// in-degree: indeg[dst] += 1 per edge
// ---------------------------------------------------------------------------
__global__ __launch_bounds__(256) void indeg_kernel(const int* __restrict__ edge_dst,
                                                    int* __restrict__ indeg, int e) {
    int i = blockIdx.x * blockDim.x + threadIdx.x;
    if (i < e) atomicAdd(&indeg[edge_dst[i]], 1);
}

// ---------------------------------------------------------------------------
// Pack a row-major [D][D] weight matrix into WMMA B-fragment order:
//   packed[(ct*16 + kq)*32 + lane] = float4{ W[kq*8+kh  ][c], W[kq*8+kh+1][c],
//                                            W[kq*8+4+kh][c], W[kq*8+5+kh][c] }
//   with kh = 2*(lane>>4), c = ct*16 + (lane&15).
// One float4 per lane feeds the B operands of two consecutive k-steps.
// grid = (16, 3*L): y selects {W_neigh, W_loop, W_evolve} x layer.
// ---------------------------------------------------------------------------
__global__ __launch_bounds__(256) void pack_kernel(
    const float* __restrict__ Wn, const float* __restrict__ Wl,
    const float* __restrict__ We, float* __restrict__ packed, int L) {
    int t = blockIdx.x * 256 + threadIdx.x;   // 0..4095 (float4 index within matrix)
    int mat = blockIdx.y;                     // 0..3L-1
    int w = mat / L, l = mat % L;
    const float* src = (w == 0 ? Wn : (w == 1 ? Wl : We)) + (size_t)l * D * D;

    int lane = t & 31;
    int kq = (t >> 5) & 15;
    int ct = t >> 9;
    int kh = (lane >> 4) << 1;
    int c = ct * 16 + (lane & 15);
    int k0 = kq * 8;

    float4 v;
    v.x = src[(k0 + kh) * D + c];
    v.y = src[(k0 + kh + 1) * D + c];
    v.z = src[(k0 + 4 + kh) * D + c];
    v.w = src[(k0 + 4 + kh + 1) * D + c];
    ((float4*)(packed + (size_t)mat * PACK_FLOATS))[t] = v;
}

// ---------------------------------------------------------------------------
// Edge message + scatter:  agg[dst] += (h[src] + rel_emb[rel]) @ Wn
// 256 threads (8 waves), 32 edges/block, each wave owns 2 16x16 C tiles.
// ---------------------------------------------------------------------------
__global__ __launch_bounds__(256) void edge_msg_kernel(
    const float* __restrict__ h, const float* __restrict__ rel_emb,
    const int* __restrict__ edge_src, const int* __restrict__ edge_rel,
    const int* __restrict__ edge_dst, const float* __restrict__ Wp,  // packed W_neigh
    float* __restrict__ agg, int E) {
    __shared__ float As[TILE * A_STRIDE];
    __shared__ int srcLds[TILE];
    __shared__ int relLds[TILE];
    __shared__ int dstLds[TILE];

    const int tid = threadIdx.x;
    const int e0 = blockIdx.x * TILE;

    if (tid < TILE) {
        int e = e0 + tid;
        bool ok = e < E;
        srcLds[tid] = ok ? edge_src[e] : 0;
        relLds[tid] = ok ? edge_rel[e] : 0;
        dstLds[tid] = ok ? edge_dst[e] : -1;
    }
    __syncthreads();

    // A = h[src] + rel_emb[rel] : 32 x 128, float4 vectorized
    for (int i = tid; i < TILE * (D / 4); i += 256) {
        int row = i >> 5;
        int c4 = i & 31;
        float4 hv = ((const float4*)(h + (size_t)srcLds[row] * D))[c4];
        float4 rv = ((const float4*)(rel_emb + (size_t)relLds[row] * D))[c4];
        float4 s;
        s.x = hv.x + rv.x; s.y = hv.y + rv.y; s.z = hv.z + rv.z; s.w = hv.w + rv.w;
        *(float4*)&As[row * A_STRIDE + c4 * 4] = s;
    }
    __syncthreads();

    const int wave = tid >> 5;
    const int lane = tid & 31;
    const int rowTile = wave >> 2;            // 0..1
    const int ct0 = (wave & 3) * 2;           // column tiles ct0, ct0+1
    const int c0 = ct0 * 16;
    const int c1 = c0 + 16;
    const int m = lane & 15;
    const int khalf = (lane >> 4) << 1;

    v8f acc0 = {};
    v8f acc1 = {};
    const float* Arow = &As[(rowTile * 16 + m) * A_STRIDE];
    const float4* B0 = (const float4*)Wp + (size_t)ct0 * 16 * 32 + lane;
    const float4* B1 = B0 + 16 * 32;
    for (int kq = 0; kq < 16; ++kq) {
        int kk = kq * 8 + khalf;
        v2f a0 = *(const v2f*)(Arow + kk);        // merged -> ds_load_2addr_b64
        v2f a1 = *(const v2f*)(Arow + kk + 4);
        float4 b0 = B0[kq * 32];                  // global_load_b128, coalesced
        float4 b1 = B1[kq * 32];
        v2f f;
        f.x = b0.x; f.y = b0.y;
        acc0 = __builtin_amdgcn_wmma_f32_16x16x4_f32(false, a0, false, f, (short)0, acc0, false, false);
        f.x = b1.x; f.y = b1.y;
        acc1 = __builtin_amdgcn_wmma_f32_16x16x4_f32(false, a0, false, f, (short)0, acc1, false, false);
        f.x = b0.z; f.y = b0.w;
        acc0 = __builtin_amdgcn_wmma_f32_16x16x4_f32(false, a1, false, f, (short)0, acc0, false, false);
        f.x = b1.z; f.y = b1.w;
        acc1 = __builtin_amdgcn_wmma_f32_16x16x4_f32(false, a1, false, f, (short)0, acc1, false, false);
    }

    // scatter C tiles: lane l VGPR r -> row (r + 8*(l>>4)), col (l&15)
    const int rbase = (lane >> 4) * 8;
    for (int r = 0; r < 8; ++r) {
        int dst = dstLds[rowTile * 16 + rbase + r];
        if (dst >= 0) {
            float* base = agg + (size_t)dst * D + m;
            unsafeAtomicAdd(base + c0, acc0[r]);   // global_atomic_add_f32
            unsafeAtomicAdd(base + c1, acc1[r]);
        }
    }
}

// ---------------------------------------------------------------------------
// Node update: h_out = rrelu(agg*norm + (indeg>0 ? h@Wl : h@We))   (in place OK)
// ---------------------------------------------------------------------------
__global__ __launch_bounds__(256) void node_update_kernel(
    const float* __restrict__ h, const float* __restrict__ agg,
    const float* __restrict__ norm, const int* __restrict__ indeg,
    const float* __restrict__ Wlp, const float* __restrict__ Wep,  // packed
    float* __restrict__ hout, int n) {
    __shared__ float As[TILE * A_STRIDE];

    const int tid = threadIdx.x;
    const int n0 = blockIdx.x * TILE;

    for (int i = tid; i < TILE * (D / 4); i += 256) {
        int row = i >> 5;
        int c4 = i & 31;
        int node = n0 + row;
        float4 v = {0.f, 0.f, 0.f, 0.f};
        if (node < n) v = ((const float4*)(h + (size_t)node * D))[c4];
        *(float4*)&As[row * A_STRIDE + c4 * 4] = v;
    }
    __syncthreads();

    const int wave = tid >> 5;
    const int lane = tid & 31;
    const int rowTile = wave >> 2;
    const int ct0 = (wave & 3) * 2;
    const int c0 = ct0 * 16;
    const int c1 = c0 + 16;
    const int m = lane & 15;
    const int khalf = (lane >> 4) << 1;

    v8f accL0 = {}, accL1 = {}, accE0 = {}, accE1 = {};
    const float* Arow = &As[(rowTile * 16 + m) * A_STRIDE];
    const float4* BL0 = (const float4*)Wlp + (size_t)ct0 * 16 * 32 + lane;
    const float4* BL1 = BL0 + 16 * 32;
    const float4* BE0 = (const float4*)Wep + (size_t)ct0 * 16 * 32 + lane;
    const float4* BE1 = BE0 + 16 * 32;
    for (int kq = 0; kq < 16; ++kq) {
        int kk = kq * 8 + khalf;
        v2f a0 = *(const v2f*)(Arow + kk);
        v2f a1 = *(const v2f*)(Arow + kk + 4);
        float4 bl0 = BL0[kq * 32];
        float4 bl1 = BL1[kq * 32];
        float4 be0 = BE0[kq * 32];
        float4 be1 = BE1[kq * 32];
        v2f f;
        f.x = bl0.x; f.y = bl0.y;
        accL0 = __builtin_amdgcn_wmma_f32_16x16x4_f32(false, a0, false, f, (short)0, accL0, false, false);
        f.x = bl1.x; f.y = bl1.y;
        accL1 = __builtin_amdgcn_wmma_f32_16x16x4_f32(false, a0, false, f, (short)0, accL1, false, false);
        f.x = be0.x; f.y = be0.y;
        accE0 = __builtin_amdgcn_wmma_f32_16x16x4_f32(false, a0, false, f, (short)0, accE0, false, false);
        f.x = be1.x; f.y = be1.y;
        accE1 = __builtin_amdgcn_wmma_f32_16x16x4_f32(false, a0, false, f, (short)0, accE1, false, false);
        f.x = bl0.z; f.y = bl0.w;
        accL0 = __builtin_amdgcn_wmma_f32_16x16x4_f32(false, a1, false, f, (short)0, accL0, false, false);
        f.x = bl1.z; f.y = bl1.w;
        accL1 = __builtin_amdgcn_wmma_f32_16x16x4_f32(false, a1, false, f, (short)0, accL1, false, false);
        f.x = be0.z; f.y = be0.w;
        accE0 = __builtin_amdgcn_wmma_f32_16x16x4_f32(false, a1, false, f, (short)0, accE0, false, false);
        f.x = be1.z; f.y = be1.w;
        accE1 = __builtin_amdgcn_wmma_f32_16x16x4_f32(false, a1, false, f, (short)0, accE1, false, false);
    }

    const int rbase = (lane >> 4) * 8;
    for (int r = 0; r < 8; ++r) {
        int node = n0 + rowTile * 16 + rbase + r;
        if (node < n) {
            float nm = norm[node];
            bool hasIn = indeg[node] > 0;
            size_t base = (size_t)node * D + m;

            float v0 = agg[base + c0] * nm + (hasIn ? accL0[r] : accE0[r]);
            v0 = (v0 >= 0.f) ? v0 : RRELU_SLOPE * v0;
            hout[base + c0] = v0;

            float v1 = agg[base + c1] * nm + (hasIn ? accL1[r] : accE1[r]);
            v1 = (v1 >= 0.f) ? v1 : RRELU_SLOPE * v1;
            hout[base + c1] = v1;
        }
    }
}

// ---------------------------------------------------------------------------
extern "C" void kernel_launch(void* const* d_in, const int* in_sizes, int n_in,
                              void* d_out, int out_size, void* d_ws, size_t ws_size,
                              hipStream_t stream) {
    const float* node_feats = (const float*)d_in[0];
    const float* rel_emb    = (const float*)d_in[1];
    const float* norm       = (const float*)d_in[2];
    const int*   edge_src   = (const int*)d_in[3];
    const int*   edge_dst   = (const int*)d_in[4];
    const int*   edge_rel   = (const int*)d_in[5];
    const float* W_neigh    = (const float*)d_in[6];
    const float* W_loop     = (const float*)d_in[7];
    const float* W_evolve   = (const float*)d_in[8];

    const int E = in_sizes[3];
    const int N = in_sizes[0] / D;
    const int L = in_sizes[6] / (D * D);

    int* indeg = (int*)d_ws;
    size_t aggOff = (((size_t)N * sizeof(int)) + 255) & ~(size_t)255;
    float* agg = (float*)((char*)d_ws + aggOff);
    size_t packOff = (aggOff + (size_t)N * D * sizeof(float) + 255) & ~(size_t)255;
    float* packed = (float*)((char*)d_ws + packOff);
    float* hout = (float*)d_out;

    hipMemsetAsync(indeg, 0, (size_t)N * sizeof(int), stream);
    indeg_kernel<<<(E + 255) / 256, 256, 0, stream>>>(edge_dst, indeg, E);
    pack_kernel<<<dim3(PACK_FLOATS / 4 / 256, 3 * L), 256, 0, stream>>>(
        W_neigh, W_loop, W_evolve, packed, L);

    const float* hin = node_feats;
    for (int l = 0; l < L; ++l) {
        hipMemsetAsync(agg, 0, (size_t)N * D * sizeof(float), stream);
        const float* pN = packed + (size_t)(0 * L + l) * PACK_FLOATS;
        const float* pL = packed + (size_t)(1 * L + l) * PACK_FLOATS;
        const float* pE = packed + (size_t)(2 * L + l) * PACK_FLOATS;
        edge_msg_kernel<<<(E + TILE - 1) / TILE, 256, 0, stream>>>(
            hin, rel_emb, edge_src, edge_rel, edge_dst, pN, agg, E);
        node_update_kernel<<<(N + TILE - 1) / TILE, 256, 0, stream>>>(
            hin, agg, norm, indeg, pL, pE, hout, N);
        hin = hout;
    }
}